// O3Linear_36094905155703
// MI455X (gfx1250) — compile-verified
//
#include <hip/hip_runtime.h>
#include <stdint.h>

typedef float v2f __attribute__((ext_vector_type(2)));
typedef float v4f __attribute__((ext_vector_type(4)));
typedef float v8f __attribute__((ext_vector_type(8)));
typedef int gv4i __attribute__((vector_size(16)));  // matches builtin's V4i
typedef __attribute__((address_space(1))) gv4i* as1_v4i_ptr;
typedef __attribute__((address_space(3))) gv4i* as3_v4i_ptr;

#define D_IN 960        // 256 + 3*128 + 5*64
#define LDS_STRIDE 964  // 960 + 4 dword pad -> conflict-free A-fragment reads

static __device__ __forceinline__ v8f wmma_f32_k4(v2f a, v2f b, v8f c) {
  // D(16x16 f32) = A(16x4 f32) x B(4x16 f32) + C
  return __builtin_amdgcn_wmma_f32_16x16x4_f32(false, a, false, b, (short)0, c,
                                               false, false);
}

template <bool FULL>
static __device__ __forceinline__ void tile_body(
    float* lds, const float* __restrict__ data_in, const float* __restrict__ W0,
    const float* __restrict__ W1, const float* __restrict__ W2,
    const float* __restrict__ b0, float* __restrict__ out, int tileRow, int n) {
  const int tid = threadIdx.x;

  // ---- Stage 16x960 activation tile into LDS (padded rows) ----
  {
    const float* src = data_in + (size_t)tileRow * D_IN;
#pragma unroll
    for (int it = 0; it < 15; ++it) {
      int idx = tid + it * 256;  // float4 index 0..3839
      int row = idx / 240;
      int c4 = idx - row * 240;
      if (FULL || tileRow + row < n) {
        const float* g = src + row * D_IN + c4 * 4;
        float* l = &lds[row * LDS_STRIDE + c4 * 4];
#if __has_builtin(__builtin_amdgcn_global_load_async_to_lds_b128)
        __builtin_amdgcn_global_load_async_to_lds_b128(
            (as1_v4i_ptr)(float*)g, (as3_v4i_ptr)l, 0, 0);
#else
        *(v4f*)l = __builtin_nontemporal_load((const v4f*)g);
#endif
      }
    }
#if __has_builtin(__builtin_amdgcn_global_load_async_to_lds_b128)
#if __has_builtin(__builtin_amdgcn_s_wait_asynccnt)
    __builtin_amdgcn_s_wait_asynccnt(0);
#else
    asm volatile("s_wait_asynccnt 0" ::: "memory");
#endif
#endif
  }
  __syncthreads();

  const int wave = tid >> 5;
  const int lane = tid & 31;
  const int hl = lane >> 4;   // half-wave select
  const int l15 = lane & 15;  // row (A) / col (B,C)
  const int kb0 = hl * 2;     // K offset within each 4-step
  const float* arow = &lds[l15 * LDS_STRIDE];
  const int orow = tileRow + hl * 8;  // C VGPR r -> output row orow + r

  // ---- GEMM0: out0 = x0 @ W0 / 16 + b0, one 16-col tile ----
  auto g0 = [&](int c0) {
    v8f acc = {};
    const float* wp = W0 + c0 + l15;
#pragma unroll 4
    for (int K = 0; K < 256; K += 4) {
      const int u0 = K + kb0;
      v2f a = *(const v2f*)(arow + u0);  // ds_load_b64, conflict-free
      v2f b;
      b.x = wp[u0 * 256];
      b.y = wp[(u0 + 1) * 256];
      acc = wmma_f32_k4(a, b, acc);
    }
    const float bias = b0[c0 + l15];
    float* op = out + (size_t)orow * D_IN + c0 + l15;
#pragma unroll
    for (int r = 0; r < 8; ++r)
      if (FULL || orow + r < n)
        __builtin_nontemporal_store(acc[r] * 0.0625f + bias, op + r * D_IN);
  };

  // ---- GEMM1: 3 channels of one w-tile, shared B fragment ----
  auto g1 = [&](int c1) {
    v8f a0 = {}, a1 = {}, a2 = {};
    const float* wp = W1 + c1 + l15;
    const float* ap = arow + 256;
#pragma unroll 2
    for (int K = 0; K < 128; K += 4) {
      const int u0 = K + kb0;
      v2f b;
      b.x = wp[u0 * 128];
      b.y = wp[(u0 + 1) * 128];
      v2f a;
      a.x = ap[u0 * 3 + 0]; a.y = ap[u0 * 3 + 3];
      a0 = wmma_f32_k4(a, b, a0);
      a.x = ap[u0 * 3 + 1]; a.y = ap[u0 * 3 + 4];
      a1 = wmma_f32_k4(a, b, a1);
      a.x = ap[u0 * 3 + 2]; a.y = ap[u0 * 3 + 5];
      a2 = wmma_f32_k4(a, b, a2);
    }
    const float s1 = 0.08838834764831843f;  // 1/sqrt(128)
    float* op = out + (size_t)orow * D_IN + 256 + (c1 + l15) * 3;
#pragma unroll
    for (int r = 0; r < 8; ++r)
      if (FULL || orow + r < n) {
        float* p = op + (size_t)r * D_IN;
        __builtin_nontemporal_store(a0[r] * s1, p + 0);
        __builtin_nontemporal_store(a1[r] * s1, p + 1);
        __builtin_nontemporal_store(a2[r] * s1, p + 2);
      }
  };

  // ---- GEMM2: 5 channels of one w-tile, shared B fragment (scale == 1) ----
  auto g2 = [&](int c2) {
    v8f acc[5] = {{}, {}, {}, {}, {}};
    const float* wp = W2 + c2 + l15;
    const float* ap = arow + 640;
#pragma unroll 2
    for (int K = 0; K < 64; K += 4) {
      const int u0 = K + kb0;
      v2f b;
      b.x = wp[u0 * 64];
      b.y = wp[(u0 + 1) * 64];
#pragma unroll
      for (int ch = 0; ch < 5; ++ch) {
        v2f a;
        a.x = ap[u0 * 5 + ch];
        a.y = ap[u0 * 5 + 5 + ch];
        acc[ch] = wmma_f32_k4(a, b, acc[ch]);
      }
    }
    float* op = out + (size_t)orow * D_IN + 640 + (c2 + l15) * 5;
#pragma unroll
    for (int r = 0; r < 8; ++r)
      if (FULL || orow + r < n) {
        float* p = op + (size_t)r * D_IN;
#pragma unroll
        for (int ch = 0; ch < 5; ++ch)
          __builtin_nontemporal_store(acc[ch][r], p + ch);
      }
  };

  // ---- Static wave -> job schedule (k-step balanced: 240 vs 288) ----
  g1(wave * 16);  // 8 w-tiles of GEMM1, one per wave
  if (wave < 4) {
    g2(wave * 16);  // 4 w-tiles of GEMM2
    g0(wave * 16);  // GEMM0 col tiles 0..3
  } else {
    const int b = 4 + (wave - 4) * 3;  // GEMM0 col tiles 4..15
    g0(b * 16);
    g0((b + 1) * 16);
    g0((b + 2) * 16);
  }
}

__global__ __launch_bounds__(256) void o3linear_kernel(
    const float* __restrict__ data_in, const float* __restrict__ W0,
    const float* __restrict__ W1, const float* __restrict__ W2,
    const float* __restrict__ b0, float* __restrict__ out, int n) {
  __shared__ float lds[16 * LDS_STRIDE];  // 61,696 B (one copy for both paths)
  const int tileRow = blockIdx.x * 16;
  if (tileRow + 16 <= n) {
    // Hot path: no per-row guards, EXEC stays all-1s, minimal SALU/VALU.
    tile_body<true>(lds, data_in, W0, W1, W2, b0, out, tileRow, n);
  } else {
    tile_body<false>(lds, data_in, W0, W1, W2, b0, out, tileRow, n);
  }
}

extern "C" void kernel_launch(void* const* d_in, const int* in_sizes, int n_in,
                              void* d_out, int out_size, void* d_ws,
                              size_t ws_size, hipStream_t stream) {
  const float* data_in = (const float*)d_in[0];
  const float* W0 = (const float*)d_in[1];
  const float* W1 = (const float*)d_in[2];
  const float* W2 = (const float*)d_in[3];
  const float* b0 = (const float*)d_in[4];
  float* out = (float*)d_out;
  const int n = in_sizes[0] / D_IN;  // 100000
  const int tiles = (n + 15) / 16;   // 6250
  hipLaunchKernelGGL(o3linear_kernel, dim3(tiles), dim3(256), 0, stream,
                     data_in, W0, W1, W2, b0, out, n);
}